// ParallelHyenaFilter_29188597744252
// MI455X (gfx1250) — compile-verified
//
#include <hip/hip_runtime.h>
#include <stdint.h>

// ---------------- problem constants ----------------
#define HIDDEN   4096
#define L_SEQ    4096
#define B_SZ     2
#define HEADS    32
#define HEAD_DIM 128
#define THREE_H  (3 * HIDDEN)
#define NSTATE   16
#define CHUNK    64
#define NCHUNK   (L_SEQ / CHUNK)
#define CT       64                 // channel tile per block
#define NCT      (HIDDEN / CT)
#define XSTR     (CT + 4)           // LDS row stride: 272B rows -> 16B aligned + bank rotate

typedef __attribute__((ext_vector_type(2))) float v2f;
typedef __attribute__((ext_vector_type(8))) float v8f;
typedef int v4i_t __attribute__((vector_size(4 * sizeof(int))));
typedef __attribute__((address_space(1))) v4i_t* g_v4i_p;
typedef __attribute__((address_space(3))) v4i_t* l_v4i_p;

// ---------------- workspace layout (float offsets) ----------------
#define T_OFF   0                               // 64x64 Toeplitz of h
#define P2_OFF  4096                            // 32x64: rows 2s=Re(pc^{63-k}), 2s+1=Im
#define R_OFF   (P2_OFF + 2048)                 // 64x32: [Re(rc*pc^{t+1}), -Im(rc*pc^{t+1})]
#define A_OFF   (R_OFF + 2048)                  // 32: pc^64 (Re/Im interleaved)
#define SL_OFF  9216                            // S_local: [b][chunk][32][4096]
#define SL_SIZE ((size_t)B_SZ * NCHUNK * 2 * NSTATE * HIDDEN)
#define CR_OFF  (SL_OFF + SL_SIZE)              // carry-in, same layout

// ---------------- CDNA5 async global->LDS path ----------------
#if defined(__AMDGCN__) && __has_builtin(__builtin_amdgcn_global_load_async_to_lds_b128)
#define HAS_ASYNC_LDS 1
#else
#define HAS_ASYNC_LDS 0
#endif

#if defined(__AMDGCN__) && __has_builtin(__builtin_amdgcn_s_wait_asynccnt)
#define HAS_WAIT_ASYNC_BUILTIN 1
#else
#define HAS_WAIT_ASYNC_BUILTIN 0
#endif

__device__ __forceinline__ void async_wait_all() {
#if HAS_ASYNC_LDS
#if HAS_WAIT_ASYNC_BUILTIN
  __builtin_amdgcn_s_wait_asynccnt(0);
#else
  asm volatile("s_wait_asynccnt 0x0" ::: "memory");
#endif
#endif
}

// copy 16 aligned bytes global -> LDS (async DMA when available)
__device__ __forceinline__ void copy16_g2l(const float* gp, float* lp) {
#if HAS_ASYNC_LDS
  __builtin_amdgcn_global_load_async_to_lds_b128(
      (g_v4i_p)(uintptr_t)gp,
      (l_v4i_p)(uint32_t)(uintptr_t)lp,
      0, 0);
#else
  *(float4*)lp = *(const float4*)gp;
#endif
}

// ================= kernel 0: build small tables =================
__global__ __launch_bounds__(256) void hyena_setup(const float* poles,
                                                   const float* residues,
                                                   float* ws) {
  __shared__ float pwre[NSTATE][CHUNK + 1];
  __shared__ float pwim[NSTATE][CHUNK + 1];
  __shared__ float hsh[CHUNK];
  int tid = threadIdx.x;
  if (tid < NSTATE) {
    float ar = poles[tid * 2 + 0], ai = poles[tid * 2 + 1];
    float pr = 1.f, pi = 0.f;
    for (int m = 0; m <= CHUNK; ++m) {
      pwre[tid][m] = pr; pwim[tid][m] = pi;
      float nr = pr * ar - pi * ai;
      float ni = pr * ai + pi * ar;
      pr = nr; pi = ni;
    }
  }
  __syncthreads();
  if (tid < CHUNK) {
    float acc = 0.f;
    for (int s = 0; s < NSTATE; ++s) {
      float rr = residues[s * 2 + 0], ri = residues[s * 2 + 1];
      acc += rr * pwre[s][tid] - ri * pwim[s][tid];   // Re(rc * pc^m)
    }
    hsh[tid] = acc;
  }
  __syncthreads();
  for (int idx = tid; idx < CHUNK * CHUNK; idx += 256) {      // T
    int r = idx / CHUNK, k = idx % CHUNK;
    ws[T_OFF + idx] = (r >= k) ? hsh[r - k] : 0.f;
  }
  for (int idx = tid; idx < 2 * NSTATE * CHUNK; idx += 256) { // P2
    int row = idx / CHUNK, k = idx % CHUNK, s = row >> 1;
    ws[P2_OFF + idx] = (row & 1) ? pwim[s][CHUNK - 1 - k] : pwre[s][CHUNK - 1 - k];
  }
  for (int idx = tid; idx < CHUNK * 2 * NSTATE; idx += 256) { // R
    int r = idx / (2 * NSTATE), j = idx % (2 * NSTATE), s = j >> 1;
    float rr = residues[s * 2 + 0], ri = residues[s * 2 + 1];
    float pr = pwre[s][r + 1], pi = pwim[s][r + 1];
    float re = rr * pr - ri * pi;
    float im = rr * pi + ri * pr;
    ws[R_OFF + idx] = (j & 1) ? -im : re;
  }
  if (tid < 2 * NSTATE) {                                     // A = pc^64
    int s = tid >> 1;
    ws[A_OFF + tid] = (tid & 1) ? pwim[s][CHUNK] : pwre[s][CHUNK];
  }
}

// ---- depthwise causal conv of one 64-channel group into LDS z ----
// stages u rows [t0-2, t0+64) via async DMA to LDS, then conv3 + bias
__device__ __forceinline__ void conv_group(const float* u, const float* w,
                                           const float* bias, int b, int t0,
                                           int ccb, int tid,
                                           float* Ush, float* Wsh, float* Bsh,
                                           float* Zsh, int mode /*0=set,1=mul*/) {
  if (tid < CT) {
    Wsh[0 * CT + tid] = w[(ccb + tid) * 3 + 0];
    Wsh[1 * CT + tid] = w[(ccb + tid) * 3 + 1];
    Wsh[2 * CT + tid] = w[(ccb + tid) * 3 + 2];
    Bsh[tid] = bias[ccb + tid];
  }
  // 66 rows x 16 transfers of 16B each
  for (int idx = tid; idx < (CHUNK + 2) * (CT / 4); idx += 256) {
    int r = idx >> 4, jj = (idx & 15) * 4;
    int t = t0 - 2 + r;
    float* lp = &Ush[r * XSTR + jj];
    if (t >= 0) {
      copy16_g2l(&u[((size_t)b * L_SEQ + t) * THREE_H + ccb + jj], lp);
    } else {
      lp[0] = 0.f; lp[1] = 0.f; lp[2] = 0.f; lp[3] = 0.f;
    }
  }
  async_wait_all();
  __syncthreads();
  for (int idx = tid; idx < CHUNK * CT; idx += 256) {
    int r = idx / CT, j = idx % CT;
    float z = Wsh[0 * CT + j] * Ush[(r)     * XSTR + j]
            + Wsh[1 * CT + j] * Ush[(r + 1) * XSTR + j]
            + Wsh[2 * CT + j] * Ush[(r + 2) * XSTR + j]
            + Bsh[j];
    if (mode == 0) Zsh[r * XSTR + j] = z;
    else           Zsh[r * XSTR + j] *= z;
  }
  __syncthreads();
}

// ========== kernel 1: conv + x1v + S_local = P2 @ X  (WMMA) ==========
__global__ __launch_bounds__(256) void hyena_chunk_state(const float* u,
                                                         const float* w,
                                                         const float* bias,
                                                         float* ws) {
  __shared__ float Ush[(CHUNK + 2) * XSTR];
  __shared__ float Xsh[CHUNK * XSTR];
  __shared__ float Wsh[3 * CT];
  __shared__ float Bsh[CT];

  int bid = blockIdx.x;
  int ct = bid & (NCT - 1);
  int chunk = (bid >> 6) & (NCHUNK - 1);
  int b = bid >> 12;
  int t0 = chunk * CHUNK;
  int c0 = ct * CT;
  int head = c0 >> 7;
  int rbase = c0 & (HEAD_DIM - 1);
  int tid = threadIdx.x;

  int cc_x1 = head * 3 * HEAD_DIM + HEAD_DIM + rbase;
  int cc_v  = head * 3 * HEAD_DIM + 2 * HEAD_DIM + rbase;
  conv_group(u, w, bias, b, t0, cc_x1, tid, Ush, Wsh, Bsh, Xsh, 0);
  conv_group(u, w, bias, b, t0, cc_v,  tid, Ush, Wsh, Bsh, Xsh, 1); // Xsh = x1*v

  // S(32x64) = P2(32x64) @ Xsh(64x64), WMMA f32 16x16x4, 8 tiles / 8 waves
  int wave = tid >> 5, lane = tid & 31, lo = lane & 15;
  int hiK = (lane < 16) ? 0 : 2;
  int tm = wave >> 2;           // 0..1
  int tn = wave & 3;            // 0..3
  const float* P2 = ws + P2_OFF;
  v8f acc = {};
  for (int kb = 0; kb < CHUNK / 4; ++kb) {
    int k0 = kb * 4;
    v2f a, bf;
    a.x = P2[(tm * 16 + lo) * CHUNK + k0 + hiK];
    a.y = P2[(tm * 16 + lo) * CHUNK + k0 + hiK + 1];
    bf.x = Xsh[(k0 + hiK)     * XSTR + tn * 16 + lo];
    bf.y = Xsh[(k0 + hiK + 1) * XSTR + tn * 16 + lo];
    acc = __builtin_amdgcn_wmma_f32_16x16x4_f32(false, a, false, bf,
                                                (short)0, acc, false, false);
  }
  size_t slbase = SL_OFF + (((size_t)b * NCHUNK + chunk) * (2 * NSTATE)) * HIDDEN;
  int madd = (lane < 16) ? 0 : 8;
  int col = c0 + tn * 16 + lo;
#pragma unroll
  for (int i = 0; i < 8; ++i) {
    int row = tm * 16 + i + madd;
    ws[slbase + (size_t)row * HIDDEN + col] = acc[i];
  }
}

// ========== kernel 2: serial scan over 64 chunks (parallel over b,s,c) ==========
__global__ __launch_bounds__(256) void hyena_scan(float* ws) {
  int tid = blockIdx.x * blockDim.x + threadIdx.x;   // B*16*4096 threads
  int c = tid & (HIDDEN - 1);
  int s = (tid >> 12) & (NSTATE - 1);
  int b = tid >> 16;
  float ar = ws[A_OFF + 2 * s], ai = ws[A_OFF + 2 * s + 1];
  float gr = 0.f, gi = 0.f;
  for (int i = 0; i < NCHUNK; ++i) {
    size_t base = SL_OFF + (((size_t)b * NCHUNK + i) * (2 * NSTATE) + 2 * s) * HIDDEN + c;
    size_t cbase = CR_OFF + (base - SL_OFF);
    ws[cbase] = gr;                 // carry-in (state before chunk i)
    ws[cbase + HIDDEN] = gi;
    float slr = ws[base], sli = ws[base + HIDDEN];
    float ngr = ar * gr - ai * gi + slr;
    float ngi = ar * gi + ai * gr + sli;
    gr = ngr; gi = ngi;
  }
}

// ========== kernel 3: Y = T@X1V + R@carry, gate, write out (WMMA) ==========
__global__ __launch_bounds__(256) void hyena_output(const float* u,
                                                    const float* w,
                                                    const float* bias,
                                                    const float* Dvec,
                                                    const float* ws,
                                                    float* out) {
  __shared__ float Ush[(CHUNK + 2) * XSTR];
  __shared__ float X1sh[CHUNK * XSTR];
  __shared__ float X2sh[CHUNK * XSTR];
  __shared__ float Csh[2 * NSTATE * XSTR];
  __shared__ float Wsh[3 * CT];
  __shared__ float Bsh[CT];

  int bid = blockIdx.x;
  int ct = bid & (NCT - 1);
  int chunk = (bid >> 6) & (NCHUNK - 1);
  int b = bid >> 12;
  int t0 = chunk * CHUNK;
  int c0 = ct * CT;
  int head = c0 >> 7;
  int rbase = c0 & (HEAD_DIM - 1);
  int tid = threadIdx.x;

  // warm L2 for the carry tile while we do the convs
  if (tid < 2 * NSTATE)
    __builtin_prefetch(&ws[CR_OFF + (((size_t)b * NCHUNK + chunk) * (2 * NSTATE) + tid) * HIDDEN + c0], 0, 0);

  int cc_x2 = head * 3 * HEAD_DIM + rbase;
  int cc_x1 = cc_x2 + HEAD_DIM;
  int cc_v  = cc_x2 + 2 * HEAD_DIM;
  conv_group(u, w, bias, b, t0, cc_x1, tid, Ush, Wsh, Bsh, X1sh, 0);
  conv_group(u, w, bias, b, t0, cc_v,  tid, Ush, Wsh, Bsh, X1sh, 1); // X1sh = x1*v
  conv_group(u, w, bias, b, t0, cc_x2, tid, Ush, Wsh, Bsh, X2sh, 0);

  // carry tile (32 x 64) via async DMA as well; rows are 16B aligned
  for (int idx = tid; idx < 2 * NSTATE * (CT / 4); idx += 256) {
    int r = idx >> 4, jj = (idx & 15) * 4;
    copy16_g2l((const float*)&ws[CR_OFF + (((size_t)b * NCHUNK + chunk) * (2 * NSTATE) + r) * HIDDEN + c0 + jj],
               &Csh[r * XSTR + jj]);
  }
  async_wait_all();
  __syncthreads();

  // Y(64x64): 16 tiles of 16x16; 8 waves x 2 tiles; K = 64 (T@X1) + 32 (R@C)
  int wave = tid >> 5, lane = tid & 31, lo = lane & 15;
  int hiK = (lane < 16) ? 0 : 2;
  int tm = wave >> 1;            // 0..3
  int tn0 = (wave & 1) * 2;      // {0,2}
  const float* Tm = ws + T_OFF;
  const float* Rm = ws + R_OFF;
  v8f acc0 = {}, acc1 = {};
  for (int kb = 0; kb < CHUNK / 4; ++kb) {
    int k0 = kb * 4;
    v2f a, b0, b1;
    a.x = Tm[(tm * 16 + lo) * CHUNK + k0 + hiK];
    a.y = Tm[(tm * 16 + lo) * CHUNK + k0 + hiK + 1];
    b0.x = X1sh[(k0 + hiK)     * XSTR + (tn0)     * 16 + lo];
    b0.y = X1sh[(k0 + hiK + 1) * XSTR + (tn0)     * 16 + lo];
    b1.x = X1sh[(k0 + hiK)     * XSTR + (tn0 + 1) * 16 + lo];
    b1.y = X1sh[(k0 + hiK + 1) * XSTR + (tn0 + 1) * 16 + lo];
    acc0 = __builtin_amdgcn_wmma_f32_16x16x4_f32(false, a, false, b0,
                                                 (short)0, acc0, false, false);
    acc1 = __builtin_amdgcn_wmma_f32_16x16x4_f32(false, a, false, b1,
                                                 (short)0, acc1, false, false);
  }
  for (int kb = 0; kb < (2 * NSTATE) / 4; ++kb) {
    int k0 = kb * 4;
    v2f a, b0, b1;
    a.x = Rm[(tm * 16 + lo) * (2 * NSTATE) + k0 + hiK];
    a.y = Rm[(tm * 16 + lo) * (2 * NSTATE) + k0 + hiK + 1];
    b0.x = Csh[(k0 + hiK)     * XSTR + (tn0)     * 16 + lo];
    b0.y = Csh[(k0 + hiK + 1) * XSTR + (tn0)     * 16 + lo];
    b1.x = Csh[(k0 + hiK)     * XSTR + (tn0 + 1) * 16 + lo];
    b1.y = Csh[(k0 + hiK + 1) * XSTR + (tn0 + 1) * 16 + lo];
    acc0 = __builtin_amdgcn_wmma_f32_16x16x4_f32(false, a, false, b0,
                                                 (short)0, acc0, false, false);
    acc1 = __builtin_amdgcn_wmma_f32_16x16x4_f32(false, a, false, b1,
                                                 (short)0, acc1, false, false);
  }

  // epilogue: out = (y + x1v*D) * x2
  int madd = (lane < 16) ? 0 : 8;
#pragma unroll
  for (int half = 0; half < 2; ++half) {
    int tn = tn0 + half;
    int col = tn * 16 + lo;
    int c = c0 + col;
    float d = Dvec[c];
    v8f acc = half ? acc1 : acc0;
#pragma unroll
    for (int i = 0; i < 8; ++i) {
      int row = tm * 16 + i + madd;
      float x1v = X1sh[row * XSTR + col];
      float x2  = X2sh[row * XSTR + col];
      out[((size_t)b * L_SEQ + t0 + row) * HIDDEN + c] = (acc[i] + x1v * d) * x2;
    }
  }
}

// ================= host launcher =================
extern "C" void kernel_launch(void* const* d_in, const int* in_sizes, int n_in,
                              void* d_out, int out_size, void* d_ws, size_t ws_size,
                              hipStream_t stream) {
  const float* u        = (const float*)d_in[0];
  const float* weight   = (const float*)d_in[1];
  const float* bias     = (const float*)d_in[2];
  const float* Dvec     = (const float*)d_in[3];
  const float* poles    = (const float*)d_in[4];
  const float* residues = (const float*)d_in[5];
  float* ws  = (float*)d_ws;
  float* out = (float*)d_out;

  hyena_setup<<<1, 256, 0, stream>>>(poles, residues, ws);
  hyena_chunk_state<<<B_SZ * NCHUNK * NCT, 256, 0, stream>>>(u, weight, bias, ws);
  hyena_scan<<<(B_SZ * NSTATE * HIDDEN) / 256, 256, 0, stream>>>(ws);
  hyena_output<<<B_SZ * NCHUNK * NCT, 256, 0, stream>>>(u, weight, bias, Dvec, ws, out);
}